// DINONewVq_51393578664434
// MI455X (gfx1250) — compile-verified
//
#include <hip/hip_runtime.h>
#include <hip/hip_bf16.h>

typedef __attribute__((ext_vector_type(2))) float v2f;
typedef __attribute__((ext_vector_type(8))) float v8f;

#define K_CB    4096
#define D_DIM   64
#define N_TOK   32768          // B*H*W = 32*32*32
#define HW      1024           // H*W
#define TILESN  (K_CB / 16)    // 256 column tiles
#define ZQ_ELEMS 2097152       // 32*64*32*32

// workspace layout (float offsets)
#define WS_ROWN  0             // 32768 row norms ||z_i||^2
#define WS_COLN  32768         // 4096  col norms ||c_j||^2
#define WS_M     36864         // 32768 softmax running max
#define WS_S     69632         // 32768 softmax sums
#define WS_IDX   102400        // 32768 argmin indices (int)
#define WS_ACC   135168        // loss accumulator

// ---------------------------------------------------------------- helpers ---

// A-matrix fragments for V_WMMA_F32_16X16X4_F32:
// lane L holds row M = L%16; VGPR0 = A[M][k0], VGPR1 = A[M][k0+1], k0 = 4f + 2*(L/16).
// z is (B, D, H, W): z_flat[n][k] = z[(b*64 + k)*1024 + hw], n = b*1024 + hw.
__device__ __forceinline__ void load_afrag(const float* __restrict__ z, int n0,
                                           int lane, v2f a[16]) {
    const int m  = lane & 15;
    const int kh = (lane >> 4) << 1;           // 0 or 2
    const int b  = n0 >> 10;
    const int hw = (n0 & 1023) + m;            // 16-block never crosses a batch
    const float* p = z + ((size_t)b * D_DIM + kh) * HW + hw;
#pragma unroll
    for (int f = 0; f < 16; ++f) {
        v2f a2;
        a2.x = p[(size_t)(4 * f) * HW];
        a2.y = p[(size_t)(4 * f + 1) * HW];
        a[f] = a2;
    }
}

// One 16x16 tile of z_flat @ codebook^T via 16 chained f32 WMMAs over D=64.
// B-matrix: lane L holds col N = L%16; VGPR pair = codebook[N][k0], [k0+1]
// (contiguous -> single b64 load per fragment).
__device__ __forceinline__ v8f tile_gemm(const v2f a[16],
                                         const float* __restrict__ cbrow) {
    v8f acc = {};
#pragma unroll
    for (int f = 0; f < 16; ++f) {
        v2f bfrag = *(const v2f*)(cbrow + 4 * f);
        acc = __builtin_amdgcn_wmma_f32_16x16x4_f32(
            /*neg_a=*/false, a[f], /*neg_b=*/false, bfrag,
            /*c_mod=*/(short)0, acc, /*reuse_a=*/false, /*reuse_b=*/false);
    }
    return acc;
}

// ---------------------------------------------------------------- kernels ---

// Pre-pass: row norms, col norms, zero loss accumulator.
__global__ void vq_pre(const float* __restrict__ z, const float* __restrict__ cb,
                       float* __restrict__ ws) {
    int i = blockIdx.x * blockDim.x + threadIdx.x;
    if (i == 0) ws[WS_ACC] = 0.0f;
    if (i < N_TOK) {
        int b = i >> 10, hw = i & 1023;
        const float* p = z + (size_t)b * D_DIM * HW + hw;
        float s = 0.0f;
#pragma unroll
        for (int dd = 0; dd < D_DIM; ++dd) {
            float v = p[(size_t)dd * HW];
            s = fmaf(v, v, s);
        }
        ws[WS_ROWN + i] = s;
    } else if (i < N_TOK + K_CB) {
        int j = i - N_TOK;
        const float4* p = (const float4*)(cb + (size_t)j * D_DIM);
        float s = 0.0f;
#pragma unroll
        for (int q = 0; q < 16; ++q) {
            float4 v = p[q];
            s = fmaf(v.x, v.x, fmaf(v.y, v.y, fmaf(v.z, v.z, fmaf(v.w, v.w, s))));
        }
        ws[WS_COLN + j] = s;
    }
}

// Pass 1: online softmax stats (max, sum of exp) + argmin per token.
// One wave per 16-token block; 8 waves / block; 256 blocks.
__global__ void __launch_bounds__(256) vq_pass1(const float* __restrict__ z,
                                                const float* __restrict__ cb,
                                                float* ws) {
    const int lane = threadIdx.x & 31;
    const int wave = threadIdx.x >> 5;
    const int n0   = (blockIdx.x * 8 + wave) * 16;
    const int half = lane >> 4;
    const int coll = lane & 15;
    const int kh   = half << 1;

    v2f a[16];
    load_afrag(z, n0, lane, a);

    float rn[8];
#pragma unroll
    for (int v = 0; v < 8; ++v) rn[v] = ws[WS_ROWN + n0 + half * 8 + v];

    float mrow[8], srow[8], dmin[8];
    int   imin[8];
#pragma unroll
    for (int v = 0; v < 8; ++v) {
        mrow[v] = -__builtin_inff();
        srow[v] = 0.0f;
        dmin[v] = __builtin_inff();
        imin[v] = 0;
    }

    for (int t = 0; t < TILESN; ++t) {
        const int Nb = t << 4;
        const float* cbrow = cb + (size_t)(Nb + coll) * D_DIM + kh;
        if (t + 1 < TILESN) __builtin_prefetch(cbrow + 16 * D_DIM, 0, 1);
        v8f acc = tile_gemm(a, cbrow);
        float cn = ws[WS_COLN + Nb + coll];
#pragma unroll
        for (int v = 0; v < 8; ++v) {
            float d = rn[v] + cn - 2.0f * acc[v];
            if (d < dmin[v]) { dmin[v] = d; imin[v] = Nb + coll; }
            float lg = -2.0f * d;               // -d / TEMP, TEMP = 0.5
            float mo = mrow[v];
            float mn = fmaxf(mo, lg);
            srow[v]  = srow[v] * __expf(mo - mn) + __expf(lg - mn);
            mrow[v]  = mn;
        }
    }

    // Butterfly reduce across the 16 lanes of each half (rows 0-7 / 8-15).
#pragma unroll
    for (int off = 8; off >= 1; off >>= 1) {
#pragma unroll
        for (int v = 0; v < 8; ++v) {
            float mo = __shfl_xor(mrow[v], off, 32);
            float so = __shfl_xor(srow[v], off, 32);
            float mn = fmaxf(mrow[v], mo);
            srow[v]  = srow[v] * __expf(mrow[v] - mn) + so * __expf(mo - mn);
            mrow[v]  = mn;
            float dob = __shfl_xor(dmin[v], off, 32);
            int   iob = __shfl_xor(imin[v], off, 32);
            if (dob < dmin[v] || (dob == dmin[v] && iob < imin[v])) {
                dmin[v] = dob; imin[v] = iob;
            }
        }
    }

    if (coll == 0) {
#pragma unroll
        for (int v = 0; v < 8; ++v) {
            int row = n0 + half * 8 + v;
            ws[WS_M + row] = mrow[v];
            ws[WS_S + row] = srow[v];
            ((int*)ws)[WS_IDX + row] = imin[v];
        }
    }
}

// Pass 2: recompute tiles, write softmax probabilities (512 MB, NT stores).
__global__ void __launch_bounds__(256) vq_pass2(const float* __restrict__ z,
                                                const float* __restrict__ cb,
                                                const float* ws,
                                                float* __restrict__ prob) {
    const int lane = threadIdx.x & 31;
    const int wave = threadIdx.x >> 5;
    const int n0   = (blockIdx.x * 8 + wave) * 16;
    const int half = lane >> 4;
    const int coll = lane & 15;
    const int kh   = half << 1;

    v2f a[16];
    load_afrag(z, n0, lane, a);

    float rn[8], mrow[8], sinv[8];
#pragma unroll
    for (int v = 0; v < 8; ++v) {
        int row = n0 + half * 8 + v;
        rn[v]   = ws[WS_ROWN + row];
        mrow[v] = ws[WS_M + row];
        sinv[v] = 1.0f / ws[WS_S + row];
    }

    for (int t = 0; t < TILESN; ++t) {
        const int Nb = t << 4;
        const float* cbrow = cb + (size_t)(Nb + coll) * D_DIM + kh;
        if (t + 1 < TILESN) __builtin_prefetch(cbrow + 16 * D_DIM, 0, 1);
        v8f acc = tile_gemm(a, cbrow);
        float cn = ws[WS_COLN + Nb + coll];
#pragma unroll
        for (int v = 0; v < 8; ++v) {
            float d = rn[v] + cn - 2.0f * acc[v];
            float p = __expf(-2.0f * d - mrow[v]) * sinv[v];
            int row = n0 + half * 8 + v;
            __builtin_nontemporal_store(p, &prob[(size_t)row * K_CB + Nb + coll]);
        }
    }
}

// Pass 3: z_q_out gather (straight-through value == codebook[idx]) + loss sum.
__global__ void __launch_bounds__(256) vq_pass3(const float* __restrict__ z,
                                                const float* __restrict__ cb,
                                                const float* ws,
                                                float* ws_acc,
                                                float* __restrict__ zq) {
    __shared__ float red[256];
    int i  = blockIdx.x * 256 + threadIdx.x;   // linear index into (B,D,H,W)
    int dd = (i >> 10) & 63;
    int b  = i >> 16;
    int hw = i & 1023;
    int n  = (b << 10) + hw;
    int idx = ((const int*)ws)[WS_IDX + n];
    float c  = cb[(size_t)idx * D_DIM + dd];
    float zv = z[i];
    zq[i] = c;
    float diff = c - zv;
    red[threadIdx.x] = diff * diff;
    __syncthreads();
#pragma unroll
    for (int s = 128; s > 0; s >>= 1) {
        if (threadIdx.x < s) red[threadIdx.x] += red[threadIdx.x + s];
        __syncthreads();
    }
    if (threadIdx.x == 0) atomicAdd(ws_acc, red[0]);
}

// q_loss = codebook_loss + 0.25*commitment_loss = 1.25 * mean(diff^2)
__global__ void vq_final(const float* __restrict__ ws, float* __restrict__ qloss) {
    *qloss = 1.25f * ws[WS_ACC] / (float)ZQ_ELEMS;
}

// ----------------------------------------------------------------- launch ---

extern "C" void kernel_launch(void* const* d_in, const int* in_sizes, int n_in,
                              void* d_out, int out_size, void* d_ws, size_t ws_size,
                              hipStream_t stream) {
    const float* z  = (const float*)d_in[0];   // (32, 64, 32, 32)
    const float* cb = (const float*)d_in[1];   // (4096, 64)
    float* ws   = (float*)d_ws;
    float* out  = (float*)d_out;
    float* zq   = out;                         // 2,097,152 floats
    float* qls  = out + ZQ_ELEMS;              // 1 float
    float* prob = out + ZQ_ELEMS + 1;          // 32768*4096 floats

    vq_pre  <<<(N_TOK + K_CB + 255) / 256, 256, 0, stream>>>(z, cb, ws);
    vq_pass1<<<256,  256, 0, stream>>>(z, cb, ws);
    vq_pass2<<<256,  256, 0, stream>>>(z, cb, ws, prob);
    vq_pass3<<<ZQ_ELEMS / 256, 256, 0, stream>>>(z, cb, ws, ws + WS_ACC, zq);
    vq_final<<<1, 1, 0, stream>>>(ws, qls);
}